// DetectionLayer_69956427317478
// MI455X (gfx1250) — compile-verified
//
#include <hip/hip_runtime.h>
#include <stdint.h>

#define NUM_ATTRS 85
#define NBOX 3
#define C_TOTAL 255          // 3 * 85 channels
#define HW 5776              // 76*76 spatial positions
#define GW 76
#define STILE 32             // spatial columns per tile (128B rows for the TDM)
#define NTILES ((HW + STILE - 1) / STILE)   // 181 (180 full + one 16-wide edge)
#define BLOCK 256
#define LDS_STRIDE_FULL 37   // 32 data + 5 pad floats -> odd stride, bank-conflict free
#define LDS_STRIDE_EDGE 21   // 16 data + 5 pad floats -> odd stride, bank-conflict free

typedef unsigned int v4u __attribute__((ext_vector_type(4)));
typedef int          v4i __attribute__((ext_vector_type(4)));
typedef int          v8i __attribute__((ext_vector_type(8)));
typedef float        v4f __attribute__((ext_vector_type(4)));

__device__ __forceinline__ float fast_exp(float x) {
  return __builtin_amdgcn_exp2f(x * 1.44269504088896340736f);   // v_exp_f32
}
__device__ __forceinline__ float fast_sigmoid(float x) {
  return __builtin_amdgcn_rcpf(1.0f + fast_exp(-x));            // v_rcp_f32
}

__global__ __launch_bounds__(BLOCK) void yolo_head_tdm(const float* __restrict__ in,
                                                       float* __restrict__ out) {
  __shared__ float smem[C_TOTAL * LDS_STRIDE_FULL];   // 37,740 B -> 8 WGs/WGP

  const int blk  = blockIdx.x;
  const int b    = blk / NTILES;
  const int tile = blk - b * NTILES;
  const int s0   = tile * STILE;
  const int tlen = (HW - s0 < STILE) ? (HW - s0) : STILE;   // 32 or 16
  const int lstride = (tlen == STILE) ? LDS_STRIDE_FULL : LDS_STRIDE_EDGE;

  // ---- TDM load: one descriptor moves the whole 255-row x tlen-float tile
  // (row stride 5776 floats) into LDS. pad_enable inserts 5 DWORDs per row so
  // the LDS row stride is odd (37/21 floats) => transposed reads hit all 64
  // banks conflict-free.
  if (threadIdx.x < 32) {            // only wave 0 issues the DMA (TDM ignores EXEC)
    const uint32_t lds_addr = (uint32_t)(uintptr_t)(&smem[0]); // low 32 bits = LDS byte addr
    const uint64_t gaddr =
        (uint64_t)(uintptr_t)(in + (size_t)b * C_TOTAL * HW + (size_t)s0);
    const uint32_t pad_interval = (tlen == STILE) ? 4u : 3u;   // 2^(code+1) dwords: 32 / 16
    const uint32_t pad_amount   = 4u;                          // code 4 => 5 dwords = 20B

    v4u g0;
    g0[0] = 1u;                                   // count = 1 (valid user descriptor)
    g0[1] = lds_addr;                             // lds_addr
    g0[2] = (uint32_t)gaddr;                      // global_addr[31:0]
    g0[3] = (uint32_t)(gaddr >> 32) | (2u << 30); // global_addr[56:32] | type=2 ("image")

    v8i g1;
    g1[0] = (int)((2u << 16)                      // data_size = 4 bytes
                | (1u << 20)                      // pad_enable
                | (pad_interval << 22)
                | (pad_amount << 25));
    g1[1] = (int)(((uint32_t)HW & 0xFFFFu) << 16);                   // tensor_dim0 lo16 (barrier addr = 0)
    g1[2] = (int)(((uint32_t)HW >> 16) | ((uint32_t)C_TOTAL << 16)); // dim0 hi | tensor_dim1 lo
    g1[3] = (int)(((uint32_t)C_TOTAL >> 16) | ((uint32_t)tlen << 16)); // dim1 hi | tile_dim0
    g1[4] = (int)(uint32_t)C_TOTAL;                                  // tile_dim1 = 255, tile_dim2 = 0
    g1[5] = (int)(uint32_t)HW;                                       // tensor_dim0_stride = 5776
    g1[6] = 0;                                                       // stride hi bits
    g1[7] = 0;

    v4i g2; g2[0] = 0; g2[1] = 0; g2[2] = 0; g2[3] = 0;
    v4i g3; g3[0] = 0; g3[1] = 0; g3[2] = 0; g3[3] = 0;

#if __has_include(<hip/amd_detail/amd_gfx1250_TDM.h>)
    v8i g4; g4[0]=0; g4[1]=0; g4[2]=0; g4[3]=0; g4[4]=0; g4[5]=0; g4[6]=0; g4[7]=0;
    __builtin_amdgcn_tensor_load_to_lds(g0, g1, g2, g3, g4, 0);
#else
    __builtin_amdgcn_tensor_load_to_lds(g0, g1, g2, g3, 0);
#endif
    __builtin_amdgcn_s_wait_tensorcnt(0);
  }
  __syncthreads();

  // ---- compute + fully coalesced float4 non-temporal streaming stores.
  // Local output index o = i*255 + c where c = box*85 + a; the global output
  // range for this tile is contiguous, so consecutive threads write consecutive
  // 16B chunks.
  const size_t out_base = ((size_t)b * (HW * NBOX) + (size_t)s0 * NBOX) * NUM_ATTRS;
  const int n_vec = (tlen * C_TOTAL) >> 2;

  for (int v = threadIdx.x; v < n_vec; v += BLOCK) {
    const int o0 = v << 2;
    float r[4];
#pragma unroll
    for (int k = 0; k < 4; ++k) {
      const int o = o0 + k;
      const int i = o / C_TOTAL;            // spatial index within tile
      const int c = o - i * C_TOTAL;        // channel 0..254
      const int box = (c >= 170) ? 2 : ((c >= 85) ? 1 : 0);
      const int a = c - box * 85;
      float res;
      if (a >= 4) {                          // confidence + class probs
        res = fast_sigmoid(smem[c * lstride + i]);
      } else {                               // box coords need (tx,tw) or (ty,th)
        const int cb  = box * 85;
        const int par = a & 1;               // 0: x/w, 1: y/h
        const float t_xy = smem[(cb + par) * lstride + i];
        const float t_wh = smem[(cb + 2 + par) * lstride + i];
        const int s  = s0 + i;
        const int gy = s / GW;
        const int gx = s - gy * GW;
        const float g = par ? (float)gy : (float)gx;
        const float xy = fast_sigmoid(t_xy) * 1.05f - 0.025f;   // XY_SCALE=1.05
        const float img_xy = (xy + g) * (1.0f / 76.0f);
        const float aw = (box == 0) ? 10.0f : ((box == 1) ? 16.0f : 33.0f);
        const float ah = (box == 0) ? 13.0f : ((box == 1) ? 30.0f : 23.0f);
        const float anc = (par ? ah : aw) * (1.0f / 608.0f);
        const float img_wh = fast_exp(t_wh) * anc;
        res = img_xy + ((a >= 2) ? 0.5f : -0.5f) * img_wh;      // min / max corner
      }
      r[k] = res;
    }
    v4f vec; vec.x = r[0]; vec.y = r[1]; vec.z = r[2]; vec.w = r[3];
    __builtin_nontemporal_store(vec, (v4f*)(out + out_base + (size_t)o0));
  }
}

extern "C" void kernel_launch(void* const* d_in, const int* in_sizes, int n_in,
                              void* d_out, int out_size, void* d_ws, size_t ws_size,
                              hipStream_t stream) {
  (void)n_in; (void)out_size; (void)d_ws; (void)ws_size;
  const float* in = (const float*)d_in[0];
  float* out = (float*)d_out;
  const int B = in_sizes[0] / (C_TOTAL * HW);   // 32 for the reference shapes
  dim3 grid((unsigned)(B * NTILES));
  hipLaunchKernelGGL(yolo_head_tdm, grid, dim3(BLOCK), 0, stream, in, out);
}